// Augment_83966610636997
// MI455X (gfx1250) — compile-verified
//
#include <hip/hip_runtime.h>
#include <hip/hip_bf16.h>
#include <math.h>

// Problem shape (fixed by the reference).
#define BATCH 32
#define HH 128
#define WW 128
#define DD 16
#define FF 8
// floats per (b,y,x) pixel = DD*FF = 128 -> 512 bytes -> 32 float4 chunks
#define NPIX_PLANE (HH * WW)            // 16384
#define NPIX_TOTAL (BATCH * NPIX_PLANE) // 524288

typedef float v4f __attribute__((ext_vector_type(4)));

// ---------------------------------------------------------------------------
// Kernel 1: composed (y,x) -> source-pixel LUT (batch independent).
// map[y*W+x] = yi*W+xi of the rotation source pixel, or -1 for zero-fill.
// Composition (reference order): rotate -> roll(sh,sw) -> flip axis2.
// ---------------------------------------------------------------------------
__global__ void build_map_kernel(const int* __restrict__ rot_deg,
                                 const int* __restrict__ shift_h,
                                 const int* __restrict__ shift_w,
                                 const int* __restrict__ flip2,
                                 int* __restrict__ map) {
  int idx = blockIdx.x * blockDim.x + threadIdx.x;
  if (idx >= NPIX_PLANE) return;
  int y = idx >> 7;
  int x = idx & (WW - 1);

  int sh = ((shift_h[0] % HH) + HH) % HH;
  int sw = ((shift_w[0] % WW) + WW) % WW;
  int xf = flip2[0] ? (WW - 1 - x) : x;
  int y2 = (y - sh + HH) % HH;
  int x2 = (xf - sw + WW) % WW;

  // Match JAX: theta/cos/sin in double, truncated to f32, math in f32,
  // round-half-to-even (rintf == jnp.round).
  double thd = (double)rot_deg[0] * 0.017453292519943295; // deg2rad
  float c = (float)cos(thd);
  float s = (float)sin(thd);
  const float yc = (HH - 1) * 0.5f;
  const float xc = (WW - 1) * 0.5f;
  float fx = (float)x2 - xc;
  float fy = (float)y2 - yc;
  float xs = c * fx + s * fy + xc;
  float ys = -s * fx + c * fy + yc;
  int xi = (int)rintf(xs);
  int yi = (int)rintf(ys);
  bool valid = (xi >= 0) && (xi < WW) && (yi >= 0) && (yi < HH);
  map[idx] = valid ? (yi * WW + xi) : -1;
}

// ---------------------------------------------------------------------------
// Kernel 2: streaming gather through the CDNA5 async global<->LDS movers.
// One wave32 moves one 512B pixel: lane i moves float4 chunk i.
// Per wave: 4 pixels double-ended through 4 private LDS buffers:
//   valid pixel : global_load_async_to_lds_b128 (ASYNCcnt)
//   zero pixel  : direct NT b128 zero store (no LDS)
//   s_wait_asynccnt 0
//   valid pixel : global_store_async_from_lds_b128 (ASYNCcnt)
// s_endpgm's implicit wait-idle drains the outstanding async stores.
// All branches on `m` are wave-uniform, so EXEC is all-ones at each issue.
// flip3 flips the D axis = per-lane chunk remap inside the 512B pixel.
// ---------------------------------------------------------------------------
__global__ __launch_bounds__(256) void
Augment_async_gather_kernel(const float* __restrict__ feat,
                            const int* __restrict__ map,
                            const int* __restrict__ flip3p,
                            float* __restrict__ out) {
  // 8 waves x 4 pixels x 512B = 16KB LDS staging.
  __shared__ __align__(16) unsigned char lbuf[8][4][512];

  const int lane = threadIdx.x & 31;
  const int w = threadIdx.x >> 5; // wave in block, 0..7
  const long long pixBase = ((long long)blockIdx.x * 8 + w) * 4;

  const int f3 = flip3p[0];
  const int srcChunk =
      f3 ? ((((DD - 1) - (lane >> 1)) << 1) | (lane & 1)) : lane;

  int mk[4];

#pragma unroll
  for (int k = 0; k < 4; ++k) {
    const long long p = pixBase + k;           // output pixel id
    const int yx = (int)(p & (NPIX_PLANE - 1)); // plane coord (wave-uniform)
    const long long b = p >> 14;               // batch
    const int m = map[yx];                     // source plane pixel or -1
    mk[k] = m;

    if (m >= 0) {
      // Per-lane 16B source address; 32 lanes cover the 512B pixel.
      unsigned long long src = (unsigned long long)(
          feat + (((b << 14) + (long long)m) << 7) + (long long)srcChunk * 4);
      // Low 32 bits of a flat shared pointer == LDS byte offset.
      unsigned ldsOff = (unsigned)(unsigned long long)&lbuf[w][k][lane * 16];
      asm volatile("global_load_async_to_lds_b128 %0, %1, off"
                   :
                   : "v"(ldsOff), "v"(src)
                   : "memory");
    } else {
      v4f z = {0.f, 0.f, 0.f, 0.f};
      __builtin_nontemporal_store(z, (v4f*)(out + (p << 7)) + lane);
    }
  }

  // All 4 async loads landed in LDS.
  asm volatile("s_wait_asynccnt 0x0" ::: "memory");

#pragma unroll
  for (int k = 0; k < 4; ++k) {
    if (mk[k] >= 0) {
      const long long p = pixBase + k;
      unsigned long long dst =
          (unsigned long long)(out + (p << 7) + (long long)lane * 4);
      unsigned ldsOff = (unsigned)(unsigned long long)&lbuf[w][k][lane * 16];
      asm volatile("global_store_async_from_lds_b128 %0, %1, off"
                   :
                   : "v"(dst), "v"(ldsOff)
                   : "memory");
    }
  }
}

extern "C" void kernel_launch(void* const* d_in, const int* in_sizes, int n_in,
                              void* d_out, int out_size, void* d_ws, size_t ws_size,
                              hipStream_t stream) {
  const float* feat = (const float*)d_in[0];
  const int* rot_deg = (const int*)d_in[1];
  const int* shift_h = (const int*)d_in[2];
  const int* shift_w = (const int*)d_in[3];
  const int* flip2   = (const int*)d_in[4];
  const int* flip3   = (const int*)d_in[5];
  float* out = (float*)d_out;

  int* map = (int*)d_ws; // 16384 * 4B = 64KB LUT in workspace

  // Kernel 1: 16384 threads build the LUT.
  build_map_kernel<<<NPIX_PLANE / 256, 256, 0, stream>>>(
      rot_deg, shift_h, shift_w, flip2, map);

  // Kernel 2: 524288 pixels, 32 pixels per 256-thread block.
  const int blocks = NPIX_TOTAL / 32; // 16384
  Augment_async_gather_kernel<<<blocks, 256, 0, stream>>>(feat, map, flip3, out);
}